// Self_Attn_59639915872760
// MI455X (gfx1250) — compile-verified
//
#include <hip/hip_runtime.h>
#include <hip/hip_bf16.h>

// ---------------------------------------------------------------------------
// Fused single-channel self-attention for x:[B=4, N=4096] (fp32).
//   q = wq*x+bq ; k = wk*x+bk ; v = wv*x+bv       (scalars)
//   energy[m,n] = q_m * k_n   -> rank-1, computed as WMMA 16x16x4 f32 tiles
//   attention   = softmax_n(energy)               (written out, 256 MB)
//   out         = gamma * (attention @ v) + x
// Stable-softmax row max is closed-form: q_m>=0 ? q_m*kmax : q_m*kmin.
// Bandwidth floor: 256 MB attention store / 23.3 TB/s ~ 11.5 us.
// ---------------------------------------------------------------------------

typedef __attribute__((ext_vector_type(2))) float v2f;
typedef __attribute__((ext_vector_type(8))) float v8f;

#define NN 4096
#define WAVES_PER_BLOCK 4
#define THREADS (WAVES_PER_BLOCK * 32)
#define ROWS_PER_WAVE 16
#define ROWS_PER_BLOCK (WAVES_PER_BLOCK * ROWS_PER_WAVE)

__global__ __launch_bounds__(THREADS) void self_attn_fused(
    const float* __restrict__ x,
    const float* __restrict__ pwq, const float* __restrict__ pbq,
    const float* __restrict__ pwk, const float* __restrict__ pbk,
    const float* __restrict__ pwv, const float* __restrict__ pbv,
    const float* __restrict__ pgamma,
    float* __restrict__ out, float* __restrict__ attn)
{
    __shared__ float ks[NN];                 // k = wk*x+bk  (16 KB)
    __shared__ float vs[NN];                 // v = wv*x+bv  (16 KB)
    __shared__ float sredmax[WAVES_PER_BLOCK];
    __shared__ float sredmin[WAVES_PER_BLOCK];

    const float wq = *pwq, bq = *pbq;
    const float wk = *pwk, bk = *pbk;
    const float wv = *pwv, bv = *pbv;
    const float gamma = *pgamma;

    const int blocksPerBatch = NN / ROWS_PER_BLOCK;       // 64
    const int b    = blockIdx.x / blocksPerBatch;
    const int mblk = blockIdx.x % blocksPerBatch;
    const int tid  = threadIdx.x;
    const int lane = tid & 31;
    const int wave = tid >> 5;

    const float* __restrict__ xb = x + (size_t)b * NN;

    // ---- fill k/v into LDS, track global min/max of k for closed-form row max
    float kmax = -INFINITY, kmin = INFINITY;
    for (int n = tid; n < NN; n += THREADS) {
        float xv = xb[n];
        float kk = wk * xv + bk;
        ks[n] = kk;
        vs[n] = wv * xv + bv;
        kmax = fmaxf(kmax, kk);
        kmin = fminf(kmin, kk);
    }
    for (int off = 16; off > 0; off >>= 1) {
        kmax = fmaxf(kmax, __shfl_xor(kmax, off, 32));
        kmin = fminf(kmin, __shfl_xor(kmin, off, 32));
    }
    if (lane == 0) { sredmax[wave] = kmax; sredmin[wave] = kmin; }
    __syncthreads();
    kmax = sredmax[0]; kmin = sredmin[0];
#pragma unroll
    for (int w = 1; w < WAVES_PER_BLOCK; ++w) {
        kmax = fmaxf(kmax, sredmax[w]);
        kmin = fminf(kmin, sredmin[w]);
    }

    // ---- this wave owns 16 attention rows: m0 .. m0+15
    const int m0   = mblk * ROWS_PER_BLOCK + wave * ROWS_PER_WAVE;
    const int half = lane >> 4;       // 0: rows 0..7 of tile, 1: rows 8..15
    const int col  = lane & 15;       // tile column

    // per-lane q for row (m0+col); duplicated across both halves
    float qv = wq * xb[m0 + col] + bq;
    float Mv = (qv >= 0.f) ? qv * kmax : qv * kmin;   // exact row max of energy

    // A matrix (16x4 f32): column K=0 holds q-tile, rest zero.
    // Layout: lanes 0-15 = M rows, VGPR0=K0, VGPR1=K1; lanes 16-31 = K2,K3.
    v2f A;
    A.x = (half == 0) ? qv : 0.f;
    A.y = 0.f;

    // Row max for the rows this lane's D registers hold: row(i) = i + 8*half
    float Mrow[8];
#pragma unroll
    for (int i = 0; i < 8; ++i)
        Mrow[i] = __shfl(Mv, i + 8 * half, 32);

    float S[8], Dot[8];
#pragma unroll
    for (int i = 0; i < 8; ++i) { S[i] = 0.f; Dot[i] = 0.f; }

    // ================= pass 1: row sums  S = sum e,  Dot = sum e*v ==========
    for (int n0 = 0; n0 < NN; n0 += 16) {
        // B matrix (4x16 f32): row K=0 holds k-tile, rest zero.
        float kv = ks[n0 + col];
        v2f Bm;
        Bm.x = (half == 0) ? kv : 0.f;
        Bm.y = 0.f;
        float vval = vs[n0 + col];
        v8f C = {};
        // D[m,n] = q_m * k_n  (16x16 tile on the matrix pipe)
        v8f D = __builtin_amdgcn_wmma_f32_16x16x4_f32(
            false, A, false, Bm, (short)0, C, false, false);
#pragma unroll
        for (int i = 0; i < 8; ++i) {
            float e = __expf(D[i] - Mrow[i]);   // <= 1, stable
            S[i]   += e;
            Dot[i] += e * vval;
        }
    }
    // reduce across the 16 lanes of each half (xor masks 8,4,2,1 stay in-half)
#pragma unroll
    for (int i = 0; i < 8; ++i) {
        float s = S[i], d = Dot[i];
        for (int off = 8; off > 0; off >>= 1) {
            s += __shfl_xor(s, off, 32);
            d += __shfl_xor(d, off, 32);
        }
        S[i] = s; Dot[i] = d;
    }
    float Sinv[8];
#pragma unroll
    for (int i = 0; i < 8; ++i) Sinv[i] = 1.f / S[i];

    // ================= pass 2: stream normalized attention ==================
    float* __restrict__ arow = attn + (size_t)b * NN * NN + (size_t)m0 * NN;
    for (int n0 = 0; n0 < NN; n0 += 16) {
        float kv = ks[n0 + col];
        v2f Bm;
        Bm.x = (half == 0) ? kv : 0.f;
        Bm.y = 0.f;
        v8f C = {};
        v8f D = __builtin_amdgcn_wmma_f32_16x16x4_f32(
            false, A, false, Bm, (short)0, C, false, false);
#pragma unroll
        for (int i = 0; i < 8; ++i) {
            float e = __expf(D[i] - Mrow[i]) * Sinv[i];
            int row = i + 8 * half;
            // lanes 0-15 write 64B of row i; lanes 16-31 write 64B of row i+8
            arow[(size_t)row * NN + (n0 + col)] = e;
        }
    }

    // ================= out = gamma * (attn @ v) + x =========================
    if (col == 0) {   // lane 0 -> rows 0..7, lane 16 -> rows 8..15
#pragma unroll
        for (int i = 0; i < 8; ++i) {
            int m = m0 + i + 8 * half;
            out[(size_t)b * NN + m] = gamma * (Dot[i] * Sinv[i]) + xb[m];
        }
    }
}

extern "C" void kernel_launch(void* const* d_in, const int* in_sizes, int n_in,
                              void* d_out, int out_size, void* d_ws, size_t ws_size,
                              hipStream_t stream) {
    const float* x  = (const float*)d_in[0];
    const float* wq = (const float*)d_in[1];
    const float* bq = (const float*)d_in[2];
    const float* wk = (const float*)d_in[3];
    const float* bk = (const float*)d_in[4];
    const float* wv = (const float*)d_in[5];
    const float* bv = (const float*)d_in[6];
    const float* gm = (const float*)d_in[7];

    const int B = in_sizes[0] / NN;                     // 4
    float* out  = (float*)d_out;                        // [B, N]
    float* attn = out + (size_t)B * NN;                 // [B, N, N]

    dim3 grid(B * (NN / ROWS_PER_BLOCK));               // 256 blocks
    dim3 block(THREADS);                                // 128 threads = 4 waves
    self_attn_fused<<<grid, block, 0, stream>>>(x, wq, bq, wk, bk, wv, bv, gm,
                                                out, attn);
}